// FP_30691836297242
// MI455X (gfx1250) — compile-verified
//
#include <hip/hip_runtime.h>
#include <math.h>

// Problem constants (from the reference)
#define NV 360
#define ND 512
#define NS 513            // samples per ray
#define NRAYS (NV * ND)   // 184320

// Bordered LDS image: 264-float row stride (interior rows 16B-aligned with a
// 4-col left border), 260 rows (2-row top/bottom border). 274,560 B < 320 KB.
#define LSTRIDE 264
#define LROWS   260
#define OFFX    4         // left border cols
#define OFFY    2         // top border rows
#define LDS_FLOATS (LSTRIDE * LROWS)      // 68640
#define LDS_BYTES  (LDS_FLOATS * 4)

#define BLOCK_THREADS 1024
#define WAVES_PER_BLOCK 32        // wave32
#define RAYS_PER_WAVE 4
#define RAYS_PER_BLOCK (WAVES_PER_BLOCK * RAYS_PER_WAVE)   // 128

// True clang vector types (HIP's float2/float4 classes don't work with
// nontemporal/async builtins).
typedef float f32x2 __attribute__((ext_vector_type(2)));
typedef float f32x4 __attribute__((ext_vector_type(4)));
typedef int   i32x4 __attribute__((ext_vector_type(4)));

__global__ __launch_bounds__(BLOCK_THREADS)
void fp_fanbeam_kernel(const float* __restrict__ img,
                       const float* __restrict__ grid,
                       const float* __restrict__ wgt,
                       float* __restrict__ out)
{
    extern __shared__ float lds[];
    const int tid = threadIdx.x;

    // ---- 1) Zero the whole bordered LDS tile (17160 float4) ----
    {
        const f32x4 z = {0.f, 0.f, 0.f, 0.f};
        #pragma unroll
        for (int k = 0; k < 17; ++k) {
            const int j = tid + k * BLOCK_THREADS;
            if (j < LDS_FLOATS / 4)
                *reinterpret_cast<f32x4*>(&lds[j * 4]) = z;
        }
    }
    __syncthreads();   // zeros visible before interior staging lands

    // ---- 2) Async-stage the 256x256 f32 image into the interior ----
    // 16384 float4 total, 16 per thread. Interior row y starts at
    // ((y+OFFY)*LSTRIDE + OFFX) floats -> 16B-aligned.
#if defined(__AMDGCN__) && __has_builtin(__builtin_amdgcn_global_load_async_to_lds_b128)
    {
        typedef __attribute__((address_space(1))) i32x4 GI4;
        typedef __attribute__((address_space(3))) i32x4 LI4;
        #pragma unroll
        for (int k = 0; k < 16; ++k) {
            const int j   = tid + k * BLOCK_THREADS;  // float4 index
            const int row = j >> 6;                   // 64 float4 per image row
            const int c4  = j & 63;
            __builtin_amdgcn_global_load_async_to_lds_b128(
                (GI4*)(img + (size_t)j * 4),
                (LI4*)(lds + (row + OFFY) * LSTRIDE + OFFX + (c4 << 2)),
                /*offset=*/0, /*cpol=*/0);
        }
#if __has_builtin(__builtin_amdgcn_s_wait_asynccnt)
        __builtin_amdgcn_s_wait_asynccnt(0);
#else
        asm volatile("s_wait_asynccnt 0" ::: "memory");
#endif
    }
#else
    {
        #pragma unroll
        for (int k = 0; k < 16; ++k) {
            const int j   = tid + k * BLOCK_THREADS;
            const int row = j >> 6;
            const int c4  = j & 63;
            const f32x4 v = reinterpret_cast<const f32x4*>(img)[j];
            *reinterpret_cast<f32x4*>(&lds[(row + OFFY) * LSTRIDE + OFFX + (c4 << 2)]) = v;
        }
    }
#endif
    __syncthreads();

    // ---- 3) One wave per ray, lane-per-sample streaming ----
    const int wave = tid >> 5;
    const int lane = tid & 31;
    const int blockBase = blockIdx.x * RAYS_PER_BLOCK;

    for (int i = 0; i < RAYS_PER_WAVE; ++i) {
        const int ray = blockBase + i * WAVES_PER_BLOCK + wave;
        if (ray >= NRAYS) break;                 // uniform per wave
        const size_t base = (size_t)ray * NS;
        const f32x2* __restrict__ g2 = reinterpret_cast<const f32x2*>(grid) + base;
        const float* __restrict__ wv = wgt + base;

        float acc = 0.0f;
        // Coalesced 256B (grid) + 128B (weight) per wave iteration.
        for (int s = lane; s < NS; s += 32) {
            const f32x2 g  = __builtin_nontemporal_load(&g2[s]);
            const float wt = __builtin_nontemporal_load(&wv[s]);

            // ix = ((gx+1)*256 - 1)/2 + border = gx*128 + 127.5 + OFF
            const float ix = fmaf(g.x, 128.0f, 127.5f + (float)OFFX);
            const float iy = fmaf(g.y, 128.0f, 127.5f + (float)OFFY);
            const float x0f = floorf(ix);
            const float y0f = floorf(iy);
            const float fx = ix - x0f;
            const float fy = iy - y0f;

            // Zero border => all 4 taps in-bounds; padding handled by zeros.
            const int idx = (int)y0f * LSTRIDE + (int)x0f;
            const float* r = lds + idx;
            const float v00 = r[0];
            const float v01 = r[1];
            const float v10 = r[LSTRIDE];
            const float v11 = r[LSTRIDE + 1];

            const float top = fmaf(fx, v01 - v00, v00);
            const float bot = fmaf(fx, v11 - v10, v10);
            acc = fmaf(wt, fmaf(fy, bot - top, top), acc);
        }

        // wave32 reduction
        #pragma unroll
        for (int off = 16; off > 0; off >>= 1)
            acc += __shfl_down(acc, off, 32);

        if (lane == 0)
            out[ray] = (acc == acc) ? acc : 0.0f;   // NaN -> 0
    }
}

extern "C" void kernel_launch(void* const* d_in, const int* in_sizes, int n_in,
                              void* d_out, int out_size, void* d_ws, size_t ws_size,
                              hipStream_t stream) {
    const float* img  = (const float*)d_in[0];   // [1,1,256,256]
    const float* grid = (const float*)d_in[1];   // [360,512,513,2]
    const float* wgt  = (const float*)d_in[2];   // [360,512,513]
    float* out = (float*)d_out;                  // [1,1,360,512]

    dim3 grd((NRAYS + RAYS_PER_BLOCK - 1) / RAYS_PER_BLOCK);  // 1440 blocks
    fp_fanbeam_kernel<<<grd, dim3(BLOCK_THREADS), LDS_BYTES, stream>>>(
        img, grid, wgt, out);
}